// SSNModel_2199023256138
// MI455X (gfx1250) — compile-verified
//
#include <hip/hip_runtime.h>
#include <math.h>

// ---------------------------------------------------------------------------
// SSN soft superpixel clustering for MI455X (gfx1250, wave32, WMMA).
//
// dist(n,k) softmax simplification: softmax_k(-(p2+c2-2dot)) == softmax_k(2dot-c2)
// -> p2 never needed.  The 16x16 (pixels x centers) dot tile over C=5 channels
// is computed with two V_WMMA_F32_16X16X4_F32 steps (channels padded 5->8),
// with -0.5*c2[k] folded into the WMMA C initializer.
// ---------------------------------------------------------------------------

typedef float v2f __attribute__((ext_vector_type(2)));
typedef float v8f __attribute__((ext_vector_type(8)));

#define B_    2
#define CH    5
#define H_    512
#define W_    512
#define NPIX  (H_ * W_)      // 262144
#define K_    256
#define NITER 10

#define WAVES        8
#define TPB          (WAVES * 32)       // 256 threads
#define PIX_PER_WAVE 16
#define PIX_PER_WG   (WAVES * PIX_PER_WAVE)  // 128
#define LDSK         257                // padded row stride (bank-conflict free)

// ---------------------------------------------------------------------------
// Center init: 32x32 block means -> centersP[b][8][K] (channels 5..7 = 0),
// c2[b][k] = sum_c center^2, and zero the accumulators.
// grid = (K_, B_), 256 threads.
// ---------------------------------------------------------------------------
__global__ void ssn_init(const float* __restrict__ x,
                         float* __restrict__ centersP,   // [B][8][K]
                         float* __restrict__ c2,         // [B][K]
                         float* __restrict__ accum)      // [B][6][K]
{
    const int k = blockIdx.x;
    const int b = blockIdx.y;
    const int t = threadIdx.x;

    __shared__ float sumS[8];
    __shared__ float meanS[8];
    if (t < 8) sumS[t] = 0.f;
    if (k < 6) accum[((size_t)b * 6 + k) * K_ + t] = 0.f;   // zero accumulators
    __syncthreads();

    const int gy = k >> 4, gx = k & 15;
    float part[CH];
#pragma unroll
    for (int c = 0; c < CH; ++c) part[c] = 0.f;

    for (int q = 0; q < 4; ++q) {
        const int idx = t + 256 * q;              // 0..1023 within 32x32 block
        const int r  = idx >> 5, cc = idx & 31;
        const int n  = (gy * 32 + r) * W_ + gx * 32 + cc;
#pragma unroll
        for (int c = 0; c < CH; ++c)
            part[c] += x[((size_t)b * CH + c) * NPIX + n];
    }
#pragma unroll
    for (int c = 0; c < CH; ++c) atomicAdd(&sumS[c], part[c]);
    __syncthreads();

    if (t < 8) {
        const float m = (t < CH) ? sumS[t] * (1.0f / 1024.0f) : 0.f;
        meanS[t] = m;
        centersP[((size_t)b * 8 + t) * K_ + k] = m;   // pad channels -> 0
    }
    __syncthreads();
    if (t == 0) {
        float s = 0.f;
#pragma unroll
        for (int c = 0; c < CH; ++c) s += meanS[c] * meanS[c];
        c2[b * K_ + k] = s;
    }
}

// ---------------------------------------------------------------------------
// Main pass: per wave 16 pixels x 256 centers.
//   - 16 center tiles, each 2x v_wmma_f32_16x16x4_f32 (C=5 padded to 8)
//   - per-pixel softmax over K=256 (lane pair via __shfl_xor across half-wave)
//   - FINAL=false: fused center-update accumulation (LDS ds_add_f32 + 1 global
//     atomicAdd per slot per workgroup)
//   - FINAL=true : transpose through LDS, coalesced float4 k-major output
// grid = (NPIX/128, B_), 256 threads.
// ---------------------------------------------------------------------------
template <bool FINAL>
__global__ __launch_bounds__(TPB) void ssn_pass(
    const float* __restrict__ x,
    const float* __restrict__ centersP,
    const float* __restrict__ c2,
    float* __restrict__ accum,
    float* __restrict__ out)
{
    __shared__ float affS[PIX_PER_WG][LDSK];          // scores -> softmax probs
    __shared__ float pixS[WAVES][8][PIX_PER_WAVE];    // pixel channels (accum)
    __shared__ float accS[6 * K_];                    // workgroup partials

    const int t    = threadIdx.x;
    const int wave = t >> 5;
    const int lane = t & 31;
    const int m    = lane & 15;     // row (pixel) / col (center) within tile
    const int hs   = lane >> 4;     // half-wave select (K-dim split for WMMA)
    const int b    = blockIdx.y;
    const int n0wg = blockIdx.x * PIX_PER_WG;
    const int n    = n0wg + wave * PIX_PER_WAVE + m;  // this lane's pixel

    const float* xb  = x + (size_t)b * CH * NPIX;
    const float* cb  = centersP + (size_t)b * 8 * K_;
    const float* c2b = c2 + b * K_;

    if (!FINAL) {
        for (int i = t; i < 6 * K_; i += TPB) accS[i] = 0.f;
    }

    // A-matrix (16 pixels x 4 channels per k-step); layout per ISA 7.12.2:
    // lanes 0-15 hold K=0,1 in VGPR0/1, lanes 16-31 hold K=2,3.
    v2f A0, A1;
    A0.x = xb[(size_t)(2 * hs + 0) * NPIX + n];
    A0.y = xb[(size_t)(2 * hs + 1) * NPIX + n];
    A1.x = (hs == 0) ? xb[(size_t)4 * NPIX + n] : 0.f;  // ch4 real, ch6 pad
    A1.y = 0.f;                                          // ch5 / ch7 pad

    if (!FINAL) {
        if (hs == 0) {
            pixS[wave][0][m] = A0.x;
            pixS[wave][1][m] = A0.y;
            pixS[wave][4][m] = A1.x;
        } else {
            pixS[wave][2][m] = A0.x;
            pixS[wave][3][m] = A0.y;
        }
    }
    __syncthreads();   // accS zeros visible; EXEC fully reconverged for WMMA

    // ---- distance tiles: D = A*B + (-0.5*c2)  =>  score = 2*D -------------
    for (int tt = 0; tt < 16; ++tt) {
        const int k = tt * 16 + m;
        v2f Bm0, Bm1;
        Bm0.x = cb[(2 * hs + 0) * K_ + k];
        Bm0.y = cb[(2 * hs + 1) * K_ + k];
        Bm1.x = cb[(4 + 2 * hs) * K_ + k];   // padded channels read as 0
        Bm1.y = cb[(5 + 2 * hs) * K_ + k];
        const float ci = -0.5f * c2b[k];
        v8f acc = {ci, ci, ci, ci, ci, ci, ci, ci};
        acc = __builtin_amdgcn_wmma_f32_16x16x4_f32(false, A0, false, Bm0,
                                                    (short)0, acc, false, false);
        acc = __builtin_amdgcn_wmma_f32_16x16x4_f32(false, A1, false, Bm1,
                                                    (short)0, acc, false, false);
#pragma unroll
        for (int r = 0; r < 8; ++r)
            affS[wave * PIX_PER_WAVE + r + 8 * hs][tt * 16 + m] = 2.0f * acc[r];
    }

    // ---- softmax over K=256 per pixel; lane pair (m, m+16) splits the row --
    {
        const int row = wave * PIX_PER_WAVE + m;
        const int kb  = hs * 128;
        float mx = -1e30f;
        for (int kk = 0; kk < 128; ++kk) mx = fmaxf(mx, affS[row][kb + kk]);
        mx = fmaxf(mx, __shfl_xor(mx, 16, 32));
        float sum = 0.f;
        for (int kk = 0; kk < 128; ++kk) {
            const float e = __expf(affS[row][kb + kk] - mx);
            affS[row][kb + kk] = e;
            sum += e;
        }
        sum += __shfl_xor(sum, 16, 32);
        const float inv = 1.0f / sum;
        for (int kk = 0; kk < 128; ++kk) affS[row][kb + kk] *= inv;
    }

    if (!FINAL) {
        // ---- fused center-update accumulation -----------------------------
        for (int j = 0; j < 8; ++j) {
            const int k = j * 32 + lane;
            float w = 0.f, s0 = 0.f, s1 = 0.f, s2 = 0.f, s3 = 0.f, s4 = 0.f;
#pragma unroll
            for (int p = 0; p < PIX_PER_WAVE; ++p) {
                const float a = affS[wave * PIX_PER_WAVE + p][k];
                w  += a;
                s0 += a * pixS[wave][0][p];
                s1 += a * pixS[wave][1][p];
                s2 += a * pixS[wave][2][p];
                s3 += a * pixS[wave][3][p];
                s4 += a * pixS[wave][4][p];
            }
            atomicAdd(&accS[0 * K_ + k], s0);
            atomicAdd(&accS[1 * K_ + k], s1);
            atomicAdd(&accS[2 * K_ + k], s2);
            atomicAdd(&accS[3 * K_ + k], s3);
            atomicAdd(&accS[4 * K_ + k], s4);
            atomicAdd(&accS[5 * K_ + k], w);
        }
        __syncthreads();
        float* ab = accum + (size_t)b * 6 * K_;
        for (int i = t; i < 6 * K_; i += TPB) atomicAdd(&ab[i], accS[i]);
    } else {
        __syncthreads();
        // out[b][k][n]: each wave stores 512B-contiguous float4 runs per k row
        float* ob = out + (size_t)b * K_ * NPIX;
        const int i0 = (t & 31) * 4;    // pixel offset within WG (0..124)
        const int kr = t >> 5;          // k row group 0..7
        for (int kk = 0; kk < 32; ++kk) {
            const int k = kr + 8 * kk;
            float4 v;
            v.x = affS[i0 + 0][k];
            v.y = affS[i0 + 1][k];
            v.z = affS[i0 + 2][k];
            v.w = affS[i0 + 3][k];
            *(float4*)&ob[(size_t)k * NPIX + n0wg + i0] = v;
        }
    }
}

// ---------------------------------------------------------------------------
// Finalize: new centers = num / (wsum + 1e-16), refresh c2, re-zero accum.
// grid = (B_), 256 threads (one per k).
// ---------------------------------------------------------------------------
__global__ void ssn_finalize(float* __restrict__ centersP,
                             float* __restrict__ c2,
                             float* __restrict__ accum)
{
    const int b = blockIdx.x;
    const int k = threadIdx.x;
    float* ab = accum + (size_t)b * 6 * K_;
    const float inv = 1.0f / (ab[5 * K_ + k] + 1e-16f);
    float s = 0.f;
#pragma unroll
    for (int c = 0; c < CH; ++c) {
        const float v = ab[c * K_ + k] * inv;
        centersP[((size_t)b * 8 + c) * K_ + k] = v;
        s += v * v;
        ab[c * K_ + k] = 0.f;
    }
    ab[5 * K_ + k] = 0.f;
    c2[b * K_ + k] = s;
}

// ---------------------------------------------------------------------------
extern "C" void kernel_launch(void* const* d_in, const int* in_sizes, int n_in,
                              void* d_out, int out_size, void* d_ws, size_t ws_size,
                              hipStream_t stream)
{
    const float* x   = (const float*)d_in[0];
    float*       out = (float*)d_out;

    // workspace carve-out (floats): centersP[2][8][256], c2[2][256], accum[2][6][256]
    float* ws       = (float*)d_ws;
    float* centersP = ws;                    // 4096
    float* c2v      = ws + 4096;             // 512
    float* accum    = ws + 4096 + 512;       // 3072

    ssn_init<<<dim3(K_, B_), 256, 0, stream>>>(x, centersP, c2v, accum);

    for (int it = 0; it < NITER; ++it) {
        ssn_pass<false><<<dim3(NPIX / PIX_PER_WG, B_), TPB, 0, stream>>>(
            x, centersP, c2v, accum, nullptr);
        ssn_finalize<<<dim3(B_), K_, 0, stream>>>(centersP, c2v, accum);
    }

    ssn_pass<true><<<dim3(NPIX / PIX_PER_WG, B_), TPB, 0, stream>>>(
        x, centersP, c2v, accum, out);
}